// GroupedTripletLoss_47949014892590
// MI455X (gfx1250) — compile-verified
//
#include <hip/hip_runtime.h>
#include <math.h>

typedef float v2f __attribute__((ext_vector_type(2)));
typedef float v8f __attribute__((ext_vector_type(8)));

#define DIM 32
#define MARGIN_F 1.0f

// ---------------------------------------------------------------------------
// Kernel 1: row-wise L2 normalize A and B, compute per-row squared norms of
// the normalized rows (matches reference's sum(A*A)) and dist_pos.
// One thread per row; D=32 fits in registers.
// ---------------------------------------------------------------------------
__global__ __launch_bounds__(256)
void gtl_prep(const float* __restrict__ A, const float* __restrict__ B,
              float* __restrict__ An, float* __restrict__ Bn,
              float* __restrict__ asq, float* __restrict__ bsq,
              float* __restrict__ dpos, int n) {
    int i = blockIdx.x * blockDim.x + threadIdx.x;
    if (i >= n) return;
    const float* a = A + (size_t)i * DIM;
    const float* b = B + (size_t)i * DIM;
    float av[DIM], bv[DIM];
    float sa = 0.f, sb = 0.f;
#pragma unroll
    for (int k = 0; k < DIM; ++k) {
        av[k] = a[k]; bv[k] = b[k];
        sa += av[k] * av[k];
        sb += bv[k] * bv[k];
    }
    float ra = 1.0f / fmaxf(sqrtf(sa), 1e-12f);
    float rb = 1.0f / fmaxf(sqrtf(sb), 1e-12f);
    float s2a = 0.f, s2b = 0.f, dp = 0.f;
#pragma unroll
    for (int k = 0; k < DIM; ++k) {
        float x = av[k] * ra;
        float y = bv[k] * rb;
        An[(size_t)i * DIM + k] = x;
        Bn[(size_t)i * DIM + k] = y;
        s2a += x * x;
        s2b += y * y;
        float d = x - y;
        dp += d * d;
    }
    asq[i] = s2a;
    bsq[i] = s2b;
    dpos[i] = sqrtf(dp);
}

// ---------------------------------------------------------------------------
// Kernel 2: each wave owns 16 rows; sweep all columns in 16-wide tiles.
// Per tile: 8 chained v_wmma_f32_16x16x4_f32 (K=32), then fused epilogue
// computing sq = asq[i] + bsq[j] - 2*dot and a masked running min per row.
// C/D layout (ISA 7.12.2): lane<16 -> rows 0..7 (VGPR v = row v), lane>=16 ->
// rows 8..15; column = lane&15.
// A layout: lane holds row M = lane&15; VGPR0/1 = K pair, +2 for lanes 16..31.
// ---------------------------------------------------------------------------
__global__ __launch_bounds__(256)
void gtl_pairmin(const float* __restrict__ An, const float* __restrict__ Bn,
                 const float* __restrict__ asq, const float* __restrict__ bsq,
                 const int* __restrict__ labels, float* __restrict__ minsq,
                 int n) {
    const int lane = threadIdx.x & 31;
    const int wave = threadIdx.x >> 5;
    const int i0 = (blockIdx.x * (blockDim.x >> 5) + wave) * 16;
    if (i0 >= n) return;                 // wave-uniform: EXEC stays all-ones

    const int half = lane >> 4;          // 0: lanes 0-15, 1: lanes 16-31
    const int l16 = lane & 15;

    // Preload A fragments for this wave's 16 rows (reused across all column tiles).
    v2f afrag[8];
    const float* arow = An + (size_t)(i0 + l16) * DIM;
#pragma unroll
    for (int kk = 0; kk < 8; ++kk)
        afrag[kk] = *(const v2f*)(arow + kk * 4 + half * 2);

    // Row-side constants for the epilogue (rows held by this lane in C).
    int rlab[8];
    float rasq[8];
#pragma unroll
    for (int v = 0; v < 8; ++v) {
        int r = i0 + half * 8 + v;
        rlab[v] = labels[r];
        rasq[v] = asq[r];
    }

    float minv[8];
#pragma unroll
    for (int v = 0; v < 8; ++v) minv[v] = __builtin_inff();

    for (int j0 = 0; j0 < n; j0 += 16) {
        const int col = j0 + l16;
        const float* brow = Bn + (size_t)col * DIM;

        v8f c = {0.f, 0.f, 0.f, 0.f, 0.f, 0.f, 0.f, 0.f};
#pragma unroll
        for (int kk = 0; kk < 8; ++kk) {
            v2f bfrag = *(const v2f*)(brow + kk * 4 + half * 2);
            c = __builtin_amdgcn_wmma_f32_16x16x4_f32(
                    /*neg_a=*/false, afrag[kk],
                    /*neg_b=*/false, bfrag,
                    /*c_mod=*/(short)0, c,
                    /*reuse_a=*/false, /*reuse_b=*/false);
        }

        const int clab = labels[col];
        const float cbsq = bsq[col];
#pragma unroll
        for (int v = 0; v < 8; ++v) {
            int r = i0 + half * 8 + v;
            float sq = rasq[v] + cbsq - 2.0f * c[v];
            bool valid = (rlab[v] == clab) && (r != col);
            minv[v] = fminf(minv[v], valid ? sq : __builtin_inff());
        }
    }

    // Min-reduce across the 16 lanes that share each row (xor stays in-half).
#pragma unroll
    for (int v = 0; v < 8; ++v) {
        float m = minv[v];
#pragma unroll
        for (int s = 1; s < 16; s <<= 1)
            m = fminf(m, __shfl_xor(m, s, 32));
        minv[v] = m;
    }
    if (l16 == 0) {
#pragma unroll
        for (int v = 0; v < 8; ++v)
            minsq[i0 + half * 8 + v] = minv[v];
    }
}

// ---------------------------------------------------------------------------
// Kernel 3: finalize. neg = g(minsq) with g(x)= x>0 ? sqrt(x) : 0 (inf -> 0,
// matching the "group of size < 2 skipped" rule), loss = relu(dpos-neg+1),
// output = mean. Single block reduction.
// ---------------------------------------------------------------------------
__global__ __launch_bounds__(256)
void gtl_finalize(const float* __restrict__ dpos, const float* __restrict__ minsq,
                  float* __restrict__ out, int n) {
    __shared__ float sdata[8];
    float acc = 0.f;
    for (int i = threadIdx.x; i < n; i += blockDim.x) {
        float ms = minsq[i];
        float neg;
        if (ms == __builtin_inff())
            neg = 0.f;                       // no valid negative -> 0
        else
            neg = (ms > 0.f) ? sqrtf(ms) : 0.f;
        float l = dpos[i] - neg + MARGIN_F;
        acc += (l > 0.f) ? l : 0.f;
    }
#pragma unroll
    for (int s = 16; s > 0; s >>= 1)
        acc += __shfl_xor(acc, s, 32);
    int w = threadIdx.x >> 5;
    if ((threadIdx.x & 31) == 0) sdata[w] = acc;
    __syncthreads();
    if (threadIdx.x == 0) {
        float t = 0.f;
        int nw = blockDim.x >> 5;
        for (int k = 0; k < nw; ++k) t += sdata[k];
        out[0] = t / (float)n;
    }
}

extern "C" void kernel_launch(void* const* d_in, const int* in_sizes, int n_in,
                              void* d_out, int out_size, void* d_ws, size_t ws_size,
                              hipStream_t stream) {
    const float* A = (const float*)d_in[0];
    const float* B = (const float*)d_in[1];
    const int* labels = (const int*)d_in[2];
    const int n = in_sizes[2];           // number of rows (labels count)

    float* ws = (float*)d_ws;
    float* An = ws;                               // n * 32
    float* Bn = An + (size_t)n * DIM;             // n * 32
    float* asq = Bn + (size_t)n * DIM;            // n
    float* bsq = asq + n;                         // n
    float* dpos = bsq + n;                        // n
    float* minsq = dpos + n;                      // n

    gtl_prep<<<(n + 255) / 256, 256, 0, stream>>>(A, B, An, Bn, asq, bsq, dpos, n);

    const int waves_per_block = 8;                // 256 threads, 8 row-tiles/block
    const int row_tiles = n / 16;
    const int blocks = (row_tiles + waves_per_block - 1) / waves_per_block;
    gtl_pairmin<<<blocks, 32 * waves_per_block, 0, stream>>>(
        An, Bn, asq, bsq, labels, minsq, n);

    gtl_finalize<<<1, 256, 0, stream>>>(dpos, minsq, (float*)d_out, n);
}